// PhysicsInformedNN_35811437314346
// MI455X (gfx1250) — compile-verified
//
#include <hip/hip_runtime.h>
#include <hip/hip_bf16.h>

typedef __attribute__((ext_vector_type(2))) float v2f;
typedef __attribute__((ext_vector_type(8))) float v8f;

#define NU_C  1.56e-05f
#define EPS_C 1e-08f
#define LDSS  132           // padded LDS row stride (floats) -> conflict-free A reads
#define NSE   (16 * LDSS)   // one stream tile: 16 points x 128(+pad) features

// ---------------------------------------------------------------------------
// Main kernel: one workgroup = 16 points, 8 waves; each wave owns a 16-neuron
// slice of the 128-wide layers and runs fp32 WMMA (16x16x4) for 5 streams:
// h (primal), hx, hy (1st tangents), hxx, hyy (2nd directional derivatives).
// ---------------------------------------------------------------------------
__global__ __launch_bounds__(256) void pinn_wmma_kernel(
    const float* __restrict__ x_,   const float* __restrict__ y_,
    const float* __restrict__ xn_,  const float* __restrict__ yn_,
    const float* __restrict__ sdf_, const float* __restrict__ g1_,
    const float* __restrict__ g2_,  const float* __restrict__ g3_,
    const float* __restrict__ u0_,  const float* __restrict__ v0_,
    const float* __restrict__ p0_,  const float* __restrict__ nut0_,
    const float* __restrict__ cn_,
    const float* __restrict__ W0, const float* __restrict__ b0,
    const float* __restrict__ W1, const float* __restrict__ b1,
    const float* __restrict__ W2, const float* __restrict__ b2,
    const float* __restrict__ W3, const float* __restrict__ b3,
    const float* __restrict__ W4, const float* __restrict__ b4,
    float* __restrict__ accum, int N)
{
    __shared__ float S[5 * NSE];        // 5 streams x (16 x 132) fp32  (~42 KB)
    __shared__ float part[4][5][16][4]; // final-layer partials [kchunk][stream][m][o]
    __shared__ float outb[5][16][4];    // final-layer results per stream
    __shared__ float red[16][8];        // per-point loss terms

    const int tid   = threadIdx.x;
    const int lane  = tid & 31;
    const int ntile = tid >> 5;       // wave id -> 16-neuron output slice
    const int m16   = lane & 15;      // row (point) for A / col (neuron) for B,C
    const int khi   = lane >> 4;      // lane half selects K pair {0,1} vs {2,3}
    const int base  = blockIdx.x * 16;

    // ---- stage inputs: h = 8 features, hx = e_x, hy = e_y, hxx = hyy = 0 ----
    if (tid < 128) {
        int m = tid >> 3, k = tid & 7;
        int gi = base + m; if (gi >= N) gi = N - 1;
        float hv;
        switch (k) {
            case 0:  hv = x_[gi];   break;
            case 1:  hv = y_[gi];   break;
            case 2:  hv = xn_[gi];  break;
            case 3:  hv = yn_[gi];  break;
            case 4:  hv = sdf_[gi]; break;
            case 5:  hv = g1_[gi];  break;
            case 6:  hv = g2_[gi];  break;
            default: hv = g3_[gi];  break;
        }
        S[0*NSE + m*LDSS + k] = hv;
        S[1*NSE + m*LDSS + k] = (k == 0) ? 1.f : 0.f;
        S[2*NSE + m*LDSS + k] = (k == 1) ? 1.f : 0.f;
        S[3*NSE + m*LDSS + k] = 0.f;
        S[4*NSE + m*LDSS + k] = 0.f;
    }
    __syncthreads();

    const float* Ws[4] = {W0, W1, W2, W3};
    const float* bs[4] = {b0, b1, b2, b3};
    const int n = ntile * 16 + m16;   // this lane's output-neuron column

#pragma unroll
    for (int l = 0; l < 4; ++l) {
        const float* __restrict__ W = Ws[l];
        const float bn   = bs[l][n];
        const int  Kdim  = (l == 0) ? 8 : 128;
        const bool first = (l == 0);

        v8f ch  = {bn, bn, bn, bn, bn, bn, bn, bn};  // bias -> every row, col n
        v8f cx  = {0.f, 0.f, 0.f, 0.f, 0.f, 0.f, 0.f, 0.f};
        v8f cy  = {0.f, 0.f, 0.f, 0.f, 0.f, 0.f, 0.f, 0.f};
        v8f cxx = {0.f, 0.f, 0.f, 0.f, 0.f, 0.f, 0.f, 0.f};
        v8f cyy = {0.f, 0.f, 0.f, 0.f, 0.f, 0.f, 0.f, 0.f};

#pragma unroll 8
        for (int k0 = 0; k0 < Kdim; k0 += 4) {
            const int kr = k0 + 2 * khi;  // lanes 0-15: K={k0,k0+1}; 16-31: {k0+2,k0+3}
            // A fragments (16x4 fp32): row = lane%16, adjacent K pair -> ds_load_2addr_b64
            v2f ah = *(const v2f*)&S[0*NSE + m16*LDSS + kr];
            v2f ax = *(const v2f*)&S[1*NSE + m16*LDSS + kr];
            v2f ay = *(const v2f*)&S[2*NSE + m16*LDSS + kr];
            // B fragment (4x16 fp32): col n fixed per lane, rows kr, kr+1
            v2f bw; bw[0] = W[kr * 128 + n]; bw[1] = W[(kr + 1) * 128 + n];

            ch = __builtin_amdgcn_wmma_f32_16x16x4_f32(false, ah, false, bw, (short)0, ch, false, false);
            cx = __builtin_amdgcn_wmma_f32_16x16x4_f32(false, ax, false, bw, (short)0, cx, false, false);
            cy = __builtin_amdgcn_wmma_f32_16x16x4_f32(false, ay, false, bw, (short)0, cy, false, false);
            if (!first) {  // layer 0 second-derivative inputs are exactly zero
                v2f axx = *(const v2f*)&S[3*NSE + m16*LDSS + kr];
                v2f ayy = *(const v2f*)&S[4*NSE + m16*LDSS + kr];
                cxx = __builtin_amdgcn_wmma_f32_16x16x4_f32(false, axx, false, bw, (short)0, cxx, false, false);
                cyy = __builtin_amdgcn_wmma_f32_16x16x4_f32(false, ayy, false, bw, (short)0, cyy, false, false);
            }
        }

        // tanh chain rule on the C tile: a=tanh(z); a'=(1-a^2)z'; a''=(1-a^2)z'' - 2a(1-a^2)z'^2
        float nh[8], nx[8], ny[8], nxx[8], nyy[8];
#pragma unroll
        for (int i = 0; i < 8; ++i) {
            float a  = tanhf(ch[i]);
            float t  = 1.f - a * a;
            float zx = cx[i], zy = cy[i];
            nh[i]  = a;
            nx[i]  = t * zx;
            ny[i]  = t * zy;
            nxx[i] = t * cxx[i] - 2.f * a * t * zx * zx;
            nyy[i] = t * cyy[i] - 2.f * a * t * zy * zy;
        }
        __syncthreads();   // all waves done READING S for this layer
#pragma unroll
        for (int i = 0; i < 8; ++i) {
            // C layout: VGPR i -> row (i + 8*laneHalf), col n
            int row = i + 8 * khi;
            S[0*NSE + row*LDSS + n] = nh[i];
            S[1*NSE + row*LDSS + n] = nx[i];
            S[2*NSE + row*LDSS + n] = ny[i];
            S[3*NSE + row*LDSS + n] = nxx[i];
            S[4*NSE + row*LDSS + n] = nyy[i];
        }
        __syncthreads();   // new activations visible to every wave
    }

    // ---- final linear layer 128 -> 4 for all 5 streams, all 256 threads ----
    // tid = {m[3:0], o[1:0], c[1:0]} : point m, output o, K-chunk c (32 K each).
    {
        const int m = tid >> 4;
        const int o = (tid >> 2) & 3;
        const int c = tid & 3;
        float zh = 0.f, zx = 0.f, zy = 0.f, zxx = 0.f, zyy = 0.f;
#pragma unroll 8
        for (int kk = 0; kk < 32; ++kk) {
            const int k = kk * 4 + c;    // lanes hit consecutive banks, o-broadcast
            const float w = W4[k * 4 + o];
            zh  += S[0*NSE + m*LDSS + k] * w;
            zx  += S[1*NSE + m*LDSS + k] * w;
            zy  += S[2*NSE + m*LDSS + k] * w;
            zxx += S[3*NSE + m*LDSS + k] * w;
            zyy += S[4*NSE + m*LDSS + k] * w;
        }
        part[c][0][m][o] = zh;  part[c][1][m][o] = zx;  part[c][2][m][o] = zy;
        part[c][3][m][o] = zxx; part[c][4][m][o] = zyy;
    }
    __syncthreads();
    if (tid < 64) {
        const int m = tid >> 2, o = tid & 3;
#pragma unroll
        for (int s = 0; s < 5; ++s) {
            float v = part[0][s][m][o] + part[1][s][m][o]
                    + part[2][s][m][o] + part[3][s][m][o];
            if (s == 0) v += b4[o];
            outb[s][m][o] = v;
        }
    }
    __syncthreads();

    // ---- physics residuals + loss terms per point ----
    if (tid < 16) {
        int gi = base + tid;
        float wv = (gi < N) ? 1.f : 0.f;
        if (gi >= N) gi = N - 1;
        float u = outb[0][tid][0], v = outb[0][tid][1], p = outb[0][tid][2], nut = outb[0][tid][3];
        float ux = outb[1][tid][0], vx = outb[1][tid][1], px = outb[1][tid][2], nutx = outb[1][tid][3];
        float uy = outb[2][tid][0], vy = outb[2][tid][1], py = outb[2][tid][2], nuty = outb[2][tid][3];
        float uxx = outb[3][tid][0], vxx = outb[3][tid][1];
        float uyy = outb[4][tid][0], vyy = outb[4][tid][1];

        float visc   = NU_C + nut;
        float tau_xx = nutx * ux + visc * uxx;
        float tau_yx = nuty * uy + visc * uyy;
        float tau_xy = nutx * vx + visc * vxx;
        float tau_yy = nuty * vy + visc * vyy;
        float uv_xy  = u * vx + v * ux + u * vy + v * uy;
        float f_u    = uv_xy + px - tau_xx - tau_yx;
        float f_v    = uv_xy + py - tau_xy - tau_yy;
        float ic     = ux + vy;

        float xnv = xn_[gi], ynv = yn_[gi];
        // coef_norm is (4,8) row-major: cn[r,c] = cn_[r*8+c]
        float bc = (u * (cn_[24] + EPS_C) + cn_[16]) * (xnv * (cn_[13] + EPS_C) + cn_[5])
                 + (v * (cn_[25] + EPS_C) + cn_[17]) * (ynv * (cn_[14] + EPS_C) + cn_[6]);

        float du = u0_[gi] - u, dv = v0_[gi] - v, dp = p0_[gi] - p, dn = nut0_[gi] - nut;
        red[tid][0] = wv * f_u * f_u;
        red[tid][1] = wv * f_v * f_v;
        red[tid][2] = wv * bc * bc;
        red[tid][3] = wv * du * du;
        red[tid][4] = wv * dv * dv;
        red[tid][5] = wv * dp * dp;
        red[tid][6] = wv * dn * dn;
        red[tid][7] = wv * ic * ic;
    }
    __syncthreads();
    if (tid < 8) {
        float s = 0.f;
        for (int m = 0; m < 16; ++m) s += red[m][tid];
        atomicAdd(&accum[tid], s);
    }
}

__global__ void pinn_zero_kernel(float* __restrict__ accum) {
    if (threadIdx.x < 8) accum[threadIdx.x] = 0.f;
}

__global__ void pinn_finalize_kernel(const float* __restrict__ accum,
                                     float* __restrict__ out, int N) {
    if (threadIdx.x == 0 && blockIdx.x == 0) {
        float inv   = 1.f / (float)N;
        float rans  = (accum[0] + accum[1]) * inv;
        float bc    = accum[2] * inv;
        float ul    = accum[3] * inv;
        float vl    = accum[4] * inv;
        float pl    = accum[5] * inv;
        float nl    = accum[6] * inv;
        float inlet = ul + vl + pl + nl;
        float icl   = accum[7] * inv;
        out[0] = rans + bc + inlet + icl;
        out[1] = rans;
        out[2] = bc;
        out[3] = inlet;
        out[4] = icl;
        out[5] = ul;
        out[6] = vl;
        out[7] = pl;
        out[8] = nl;
    }
}

extern "C" void kernel_launch(void* const* d_in, const int* in_sizes, int n_in,
                              void* d_out, int out_size, void* d_ws, size_t ws_size,
                              hipStream_t stream) {
    // setup_inputs() dict order:
    const float* x_    = (const float*)d_in[0];
    const float* y_    = (const float*)d_in[1];
    const float* xn_   = (const float*)d_in[2];
    const float* yn_   = (const float*)d_in[3];
    const float* sdf_  = (const float*)d_in[4];
    const float* g1_   = (const float*)d_in[5];
    const float* g2_   = (const float*)d_in[6];
    const float* g3_   = (const float*)d_in[7];
    const float* u0_   = (const float*)d_in[8];
    const float* v0_   = (const float*)d_in[9];
    const float* p0_   = (const float*)d_in[10];
    const float* nut0_ = (const float*)d_in[11];
    const float* cn_   = (const float*)d_in[12];
    const float* W0 = (const float*)d_in[13]; const float* b0 = (const float*)d_in[14];
    const float* W1 = (const float*)d_in[15]; const float* b1 = (const float*)d_in[16];
    const float* W2 = (const float*)d_in[17]; const float* b2 = (const float*)d_in[18];
    const float* W3 = (const float*)d_in[19]; const float* b3 = (const float*)d_in[20];
    const float* W4 = (const float*)d_in[21]; const float* b4 = (const float*)d_in[22];

    const int N = in_sizes[0];
    float* accum = (float*)d_ws;

    pinn_zero_kernel<<<1, 32, 0, stream>>>(accum);
    const int nblocks = (N + 15) / 16;
    pinn_wmma_kernel<<<nblocks, 256, 0, stream>>>(
        x_, y_, xn_, yn_, sdf_, g1_, g2_, g3_, u0_, v0_, p0_, nut0_, cn_,
        W0, b0, W1, b1, W2, b2, W3, b3, W4, b4, accum, N);
    pinn_finalize_kernel<<<1, 32, 0, stream>>>(accum, (float*)d_out, N);
}